// SpatialTransform_21019569946721
// MI455X (gfx1250) — compile-verified
//
#include <hip/hip_runtime.h>

// Problem constants (from reference setup_inputs):
#define BB 16
#define CC 4
#define HH 512
#define WW 512
#define HW (HH * WW)

typedef float f32x4 __attribute__((ext_vector_type(4)));
typedef float f32x2 __attribute__((ext_vector_type(2)));

// One block = two (b,h) image rows: 256 threads x 4 pixels = 1024 = 2*W.
// Memory-bound gather kernel. Inputs (96 MB) are L2-resident on MI455X
// (192 MB L2); outputs are streamed with non-temporal stores so they never
// evict the inputs. Corner pairs are fetched as single b64 loads with a
// border-select, halving the gather request count.
__global__ __launch_bounds__(256)
void SpatialTransform_21019569946721_kernel(
    const float* __restrict__ img,       // [B,C,H,W]
    const float* __restrict__ flow,      // [B,H,W,2]
    float*       __restrict__ out_grid,  // [B,H,W,2]  (output 0)
    float*       __restrict__ out_warp)  // [B,C,H,W]  (output 1)
{
    const int t   = threadIdx.x;
    const int row = blockIdx.x * 2 + (t >> 7);   // 0 .. B*H-1
    const int b   = row >> 9;                    // row / 512
    const int h   = row & (HH - 1);              // row % 512
    const int w   = (t & 127) << 2;              // 4 pixels per thread, w % 4 == 0

    // linspace(-1, 1, N): v_i = -1 + i * 2/(N-1). H == W so one scale.
    const float scale = 2.0f / (float)(WW - 1);
    const float gh = __fmaf_rn((float)h, scale, -1.0f);

    const size_t pix = (size_t)row * WW + (size_t)w;  // flat [B,H,W] pixel idx

    // Two 128-bit coalesced flow loads: (f_h, f_w) x 4 pixels
    const f32x4 f01 = *(const f32x4*)(flow + pix * 2);
    const f32x4 f23 = *(const f32x4*)(flow + pix * 2 + 4);

    // Speculative L2 prefetch of the flow stream (gfx1250 global_prefetch_b8)
    __builtin_prefetch(flow + pix * 2 + 4 * WW, 0, 1);

    // disp_h = grid_h + flow[...,0]; disp_w = grid_w + flow[...,1]
    float dh[4], dw[4];
    dh[0] = gh + f01.x;  dw[0] = __fmaf_rn((float)(w + 0), scale, -1.0f) + f01.y;
    dh[1] = gh + f01.z;  dw[1] = __fmaf_rn((float)(w + 1), scale, -1.0f) + f01.w;
    dh[2] = gh + f23.x;  dw[2] = __fmaf_rn((float)(w + 2), scale, -1.0f) + f23.y;
    dh[3] = gh + f23.z;  dw[3] = __fmaf_rn((float)(w + 3), scale, -1.0f) + f23.w;

    // Output 0: sample_grid[..., (disp_w, disp_h)] — streaming NT b128 stores
    {
        f32x4 g0, g1;
        g0.x = dw[0]; g0.y = dh[0]; g0.z = dw[1]; g0.w = dh[1];
        g1.x = dw[2]; g1.y = dh[2]; g1.z = dw[3]; g1.w = dh[3];
        __builtin_nontemporal_store(g0, (f32x4*)(out_grid + pix * 2));
        __builtin_nontemporal_store(g1, (f32x4*)(out_grid + pix * 2 + 4));
    }

    // Unnormalize + border clamp:  t = clamp((d+1)*0.5*(N-1), 0, N-1)
    const float half_span = 0.5f * (float)(WW - 1);
    const float lim = (float)(WW - 1);

    int   offT[4], offB[4];   // element offsets of top / bottom corner pairs
    float wx[4], wy[4];
    bool  edge[4];            // x0 == W-1: wx==0, pair's 2nd element is v00

#pragma unroll
    for (int i = 0; i < 4; ++i) {
        const float x  = fminf(fmaxf((dw[i] + 1.0f) * half_span, 0.0f), lim);
        const float y  = fminf(fmaxf((dh[i] + 1.0f) * half_span, 0.0f), lim);
        const float xf = floorf(x), yf = floorf(y);
        wx[i] = x - xf;
        wy[i] = y - yf;
        const int x0 = (int)xf, y0 = (int)yf;
        const int y1 = min(y0 + 1, HH - 1);
        const int xc = min(x0, WW - 2);       // pair base stays in-row
        edge[i] = (x0 == WW - 1);
        offT[i] = y0 * WW + xc;
        offB[i] = y1 * WW + xc;
    }

    const float* __restrict__ imb  = img + (size_t)b * CC * HW;
    float*       __restrict__ wout = out_warp + (size_t)b * CC * HW
                                   + (size_t)h * WW + (size_t)w;

#pragma unroll
    for (int c = 0; c < CC; ++c) {
        const float* __restrict__ p = imb + (size_t)c * HW;  // 1MB step: fits ioffset
        float res[4];
#pragma unroll
        for (int i = 0; i < 4; ++i) {
            // Two b64 corner-pair gathers (align 4 — native on gfx1250)
            f32x2 qt, qb;
            __builtin_memcpy(&qt, p + offT[i], sizeof(qt));
            __builtin_memcpy(&qb, p + offB[i], sizeof(qb));

            const float v00 = edge[i] ? qt.y : qt.x;
            const float v01 = qt.y;
            const float v10 = edge[i] ? qb.y : qb.x;
            const float v11 = qb.y;

            // Bilinear via 3 FMAs (lerp form)
            const float top = __fmaf_rn(wx[i], v01 - v00, v00);
            const float bot = __fmaf_rn(wx[i], v11 - v10, v10);
            res[i] = __fmaf_rn(wy[i], bot - top, top);
        }
        f32x4 r;
        r.x = res[0]; r.y = res[1]; r.z = res[2]; r.w = res[3];
        __builtin_nontemporal_store(r, (f32x4*)(wout + (size_t)c * HW));
    }
}

extern "C" void kernel_launch(void* const* d_in, const int* in_sizes, int n_in,
                              void* d_out, int out_size, void* d_ws, size_t ws_size,
                              hipStream_t stream) {
    (void)in_sizes; (void)n_in; (void)out_size; (void)d_ws; (void)ws_size;

    const float* img  = (const float*)d_in[0];   // mov_image [16,4,512,512]
    const float* flow = (const float*)d_in[1];   // flow      [16,512,512,2]

    // d_out = sample_grid (B*H*W*2 floats) || warped (B*C*H*W floats)
    float* out_grid = (float*)d_out;
    float* out_warp = out_grid + (size_t)BB * HH * WW * 2;

    dim3 grid(BB * HH / 2);  // 4096 blocks, two image rows per block
    dim3 block(256);         // 8 wave32 per block; 4 pixels per thread

    SpatialTransform_21019569946721_kernel<<<grid, block, 0, stream>>>(
        img, flow, out_grid, out_warp);
}